// BasicGCNSegmentation_38766374813979
// MI455X (gfx1250) — compile-verified
//
#include <hip/hip_runtime.h>

typedef __attribute__((ext_vector_type(2))) float v2f;
typedef __attribute__((ext_vector_type(8))) float v8f;

// ---------------------------------------------------------------- utilities

__global__ void k_zero(float* __restrict__ p, int n) {
    int i = blockIdx.x * blockDim.x + threadIdx.x;
    if (i < n) p[i] = 0.0f;
}

__global__ void k_degree(const int* __restrict__ src, const int* __restrict__ dst,
                         float* __restrict__ degO, float* __restrict__ degI, int e) {
    int i = blockIdx.x * blockDim.x + threadIdx.x;
    if (i < e) {
        atomicAdd(&degO[src[i]], 1.0f);
        atomicAdd(&degI[dst[i]], 1.0f);
    }
}

__global__ void k_coeff(float* __restrict__ c, int n) {
    int i = blockIdx.x * blockDim.x + threadIdx.x;
    if (i < n) {
        float d = c[i];
        c[i] = (d > 0.0f) ? rsqrtf(d) : 0.0f;   // deg>=1 when >0, matches max(deg,1)
    }
}

// features N x 6  ->  padded N x 8 (cols 6,7 = 0)
__global__ void k_pad_features(const float* __restrict__ x, float* __restrict__ xp, int n) {
    int i = blockIdx.x * blockDim.x + threadIdx.x;
    if (i < n * 8) {
        int r = i >> 3, c = i & 7;
        xp[i] = (c < 6) ? x[r * 6 + c] : 0.0f;
    }
}

// W1 6x128 -> 8x128 (rows 6,7 = 0); row-major index identical for r<6
__global__ void k_pad_w1(const float* __restrict__ w, float* __restrict__ wp) {
    int i = blockIdx.x * blockDim.x + threadIdx.x;
    if (i < 8 * 128) {
        int r = i >> 7;
        wp[i] = (r < 6) ? w[i] : 0.0f;
    }
}

// W5 128x18 -> 128x32 (cols 18..31 = 0)
__global__ void k_pad_w5(const float* __restrict__ w, float* __restrict__ wp) {
    int i = blockIdx.x * blockDim.x + threadIdx.x;
    if (i < 128 * 32) {
        int r = i >> 5, c = i & 31;
        wp[i] = (c < 18) ? w[r * 18 + c] : 0.0f;
    }
}

// ---------------------------------------------------------------- WMMA GEMM
// Y[N x DOUT] = A[N x DIN] @ W[DIN x DOUT], full fp32 via V_WMMA_F32_16X16X4_F32.
// One wave (blockDim=32) computes one 16x16 tile; grid = (N/16, DOUT/16).
// A frag 16x4: lanes 0-15 hold M=lane, K=k+0/k+1 (v0/v1); lanes 16-31 hold K=k+2/k+3.
// B frag 4x16: lanes 0-15 hold N=lane, K=k+0/k+1; lanes 16-31 K=k+2/k+3.
// C/D 16x16: VGPR v -> M=v (lanes 0-15) / M=v+8 (lanes 16-31), N=lane&15.
template <int DIN, int DOUT>
__global__ void k_gemm_wmma(const float* __restrict__ A, const float* __restrict__ W,
                            float* __restrict__ Y) {
    const int lane = threadIdx.x;
    const int m0   = blockIdx.x * 16;
    const int n0   = blockIdx.y * 16;
    const int half = lane >> 4;          // 0 or 1
    const int l15  = lane & 15;
    const int koff = half * 2;

    const float* arow = A + (size_t)(m0 + l15) * DIN + koff;
    const int col = n0 + l15;

    v8f acc = {};
#pragma unroll
    for (int k = 0; k < DIN; k += 4) {
        v2f a = *(const v2f*)(arow + k);                  // 8B-aligned (even index)
        v2f b;
        b.x = W[(size_t)(k + koff) * DOUT + col];
        b.y = W[(size_t)(k + koff + 1) * DOUT + col];
        acc = __builtin_amdgcn_wmma_f32_16x16x4_f32(
            /*neg_a=*/false, a, /*neg_b=*/false, b,
            /*c_mod=*/(short)0, acc, /*reuse_a=*/false, /*reuse_b=*/false);
    }

    float* yrow = Y + (size_t)(m0 + half * 8) * DOUT + col;
#pragma unroll
    for (int v = 0; v < 8; ++v) yrow[(size_t)v * DOUT] = acc[v];
}

// ------------------------------------------------------------- edge scatter
// One wave per edge: coalesced row load from XW[src], scaled, float-atomicAdd
// into AGG[dst] (global_atomic_add_f32).
template <int DOUT, int STRIDE>
__global__ void k_scatter(const float* __restrict__ XW, const int* __restrict__ src,
                          const int* __restrict__ dst, const float* __restrict__ cs,
                          float* __restrict__ AGG, int e) {
    int w    = (blockIdx.x * blockDim.x + threadIdx.x) >> 5;
    int lane = threadIdx.x & 31;
    if (w >= e) return;
    int s = src[w], d = dst[w];
    float c = cs[s];
    const float* xr = XW + (size_t)s * STRIDE;
    float* ar       = AGG + (size_t)d * STRIDE;
#pragma unroll
    for (int j = lane; j < DOUT; j += 32)
        atomicAdd(&ar[j], xr[j] * c);
}

// ------------------------------------------------- scale + bias (+ReLU), in place
template <int DOUT, int STRIDE, bool RELU>
__global__ void k_finalize(float* __restrict__ H, const float* __restrict__ cd,
                           const float* __restrict__ b, int n) {
    int i = blockIdx.x * blockDim.x + threadIdx.x;
    if (i < n * DOUT) {
        int r = i / DOUT, c = i % DOUT;
        size_t idx = (size_t)r * STRIDE + c;
        float v = H[idx] * cd[r] + b[c];
        if (RELU) v = fmaxf(v, 0.0f);
        H[idx] = v;
    }
}

// final logits: AGG (stride 32) -> out (N x 18, compact), no ReLU
__global__ void k_finalize_out(const float* __restrict__ AGG, const float* __restrict__ cd,
                               const float* __restrict__ b, float* __restrict__ out, int n) {
    int i = blockIdx.x * blockDim.x + threadIdx.x;
    if (i < n * 18) {
        int r = i / 18, c = i % 18;
        out[i] = AGG[(size_t)r * 32 + c] * cd[r] + b[c];
    }
}

// ---------------------------------------------------------------- launcher

static inline int cdiv(int a, int b) { return (a + b - 1) / b; }

extern "C" void kernel_launch(void* const* d_in, const int* in_sizes, int n_in,
                              void* d_out, int out_size, void* d_ws, size_t ws_size,
                              hipStream_t stream) {
    (void)n_in; (void)out_size; (void)ws_size;

    const float* feat = (const float*)d_in[0];
    const int*   src  = (const int*)d_in[1];
    const int*   dst  = (const int*)d_in[2];
    const float* W1 = (const float*)d_in[3];  const float* b1 = (const float*)d_in[4];
    const float* W2 = (const float*)d_in[5];  const float* b2 = (const float*)d_in[6];
    const float* W3 = (const float*)d_in[7];  const float* b3 = (const float*)d_in[8];
    const float* W4 = (const float*)d_in[9];  const float* b4 = (const float*)d_in[10];
    const float* W5 = (const float*)d_in[11]; const float* b5 = (const float*)d_in[12];
    float* out = (float*)d_out;

    const int N = 100000;           // divisible by 16
    const int E = in_sizes[1];      // 1.6M

    float* ws = (float*)d_ws;
    size_t o = 0;
    float* c_src = ws + o; o += N;
    float* c_dst = ws + o; o += N;
    float* Xp    = ws + o; o += (size_t)N * 8;
    float* W1p   = ws + o; o += 8 * 128;
    float* W5p   = ws + o; o += 128 * 32;
    float* XW    = ws + o; o += (size_t)N * 128;
    float* H0    = ws + o; o += (size_t)N * 128;
    float* H1    = ws + o; o += (size_t)N * 128;

    // --- normalizers ---
    k_zero<<<cdiv(2 * N, 256), 256, 0, stream>>>(c_src, 2 * N);   // c_src,c_dst contiguous
    k_degree<<<cdiv(E, 256), 256, 0, stream>>>(src, dst, c_src, c_dst, E);
    k_coeff<<<cdiv(2 * N, 256), 256, 0, stream>>>(c_src, 2 * N);

    // --- padding ---
    k_pad_features<<<cdiv(N * 8, 256), 256, 0, stream>>>(feat, Xp, N);
    k_pad_w1<<<4, 256, 0, stream>>>(W1, W1p);
    k_pad_w5<<<16, 256, 0, stream>>>(W5, W5p);

    const dim3 gTile(N / 16, 128 / 16);   // 6250 x 8
    const int  sBlk = cdiv(E, 8);         // 8 edges per 256-thread block
    const int  fBlk = cdiv(N * 128, 256);

    // --- layer 1: Xp(8) @ W1p -> H0, ReLU ---
    k_gemm_wmma<8, 128><<<gTile, 32, 0, stream>>>(Xp, W1p, XW);
    k_zero<<<fBlk, 256, 0, stream>>>(H0, N * 128);
    k_scatter<128, 128><<<sBlk, 256, 0, stream>>>(XW, src, dst, c_src, H0, E);
    k_finalize<128, 128, true><<<fBlk, 256, 0, stream>>>(H0, c_dst, b1, N);

    // --- layer 2: H0 -> H1 ---
    k_gemm_wmma<128, 128><<<gTile, 32, 0, stream>>>(H0, W2, XW);
    k_zero<<<fBlk, 256, 0, stream>>>(H1, N * 128);
    k_scatter<128, 128><<<sBlk, 256, 0, stream>>>(XW, src, dst, c_src, H1, E);
    k_finalize<128, 128, true><<<fBlk, 256, 0, stream>>>(H1, c_dst, b2, N);

    // --- layer 3: H1 -> H0 ---
    k_gemm_wmma<128, 128><<<gTile, 32, 0, stream>>>(H1, W3, XW);
    k_zero<<<fBlk, 256, 0, stream>>>(H0, N * 128);
    k_scatter<128, 128><<<sBlk, 256, 0, stream>>>(XW, src, dst, c_src, H0, E);
    k_finalize<128, 128, true><<<fBlk, 256, 0, stream>>>(H0, c_dst, b3, N);

    // --- layer 4: H0 -> H1 ---
    k_gemm_wmma<128, 128><<<gTile, 32, 0, stream>>>(H0, W4, XW);
    k_zero<<<fBlk, 256, 0, stream>>>(H1, N * 128);
    k_scatter<128, 128><<<sBlk, 256, 0, stream>>>(XW, src, dst, c_src, H1, E);
    k_finalize<128, 128, true><<<fBlk, 256, 0, stream>>>(H1, c_dst, b4, N);

    // --- layer 5: H1(128) @ W5p(128x32) -> logits (18 cols live) ---
    const dim3 gTile5(N / 16, 32 / 16);
    k_gemm_wmma<128, 32><<<gTile5, 32, 0, stream>>>(H1, W5p, XW);   // XW used as N x 32
    k_zero<<<cdiv(N * 32, 256), 256, 0, stream>>>(H0, N * 32);      // reuse H0 as AGG
    k_scatter<18, 32><<<sBlk, 256, 0, stream>>>(XW, src, dst, c_src, H0, E);
    k_finalize_out<<<cdiv(N * 18, 256), 256, 0, stream>>>(H0, c_dst, b5, out, N);
}